// GCNLayer_13434657702012
// MI455X (gfx1250) — compile-verified
//
#include <hip/hip_runtime.h>

typedef __attribute__((ext_vector_type(2))) float v2f;
typedef __attribute__((ext_vector_type(8))) float v8f;

#define IN_DIM 256
#define OUT_DIM 256

// ---------------------------------------------------------------------------
// Phase 0: zero the aggregation buffer (d_ws is poisoned by the harness).
// b128 stores; aggElems (N*256) is divisible by 4.
// ---------------------------------------------------------------------------
__global__ void GCN_zero_kernel(float4* __restrict__ p, long long n4) {
  long long i = (long long)blockIdx.x * blockDim.x + threadIdx.x;
  long long stride = (long long)gridDim.x * blockDim.x;
  float4 z = {0.f, 0.f, 0.f, 0.f};
  for (; i < n4; i += stride) p[i] = z;
}

// ---------------------------------------------------------------------------
// Phase 1: COO SpMM scatter.  agg[row[e], :] += vals[e] * x[col[e], :]
// One wave32 per edge (8 edges per 256-thread block); lane L handles
// features L, L+32, ..., L+224. Gather and atomic scatter are coalesced
// 128B segments; x (102MB) and agg (102MB) both stay resident in the
// 192MB L2, so this phase is L2-atomic-bandwidth bound.
// ---------------------------------------------------------------------------
__global__ void GCN_spmm_scatter_kernel(const int* __restrict__ row,
                                        const int* __restrict__ col,
                                        const float* __restrict__ vals,
                                        const float* __restrict__ x,
                                        float* __restrict__ agg,
                                        int E) {
  const int wave = (int)((blockIdx.x * blockDim.x + threadIdx.x) >> 5);
  const int lane = threadIdx.x & 31;
  if (wave >= E) return;                 // uniform per wave

  const int r = row[wave];
  const int c = col[wave];
  const float v = vals[wave];

  const float* xr = x   + (long long)c * IN_DIM + lane;
  float*       ar = agg + (long long)r * IN_DIM + lane;

  #pragma unroll
  for (int i = 0; i < IN_DIM / 32; ++i) {
    atomicAdd(ar + i * 32, v * xr[i * 32]);
  }
}

// ---------------------------------------------------------------------------
// Phase 2: out = agg @ W + b via V_WMMA_F32_16X16X4_F32 (full-precision fp32
// matrix pipe, matching the reference dtype).
// Register blocking: one wave computes a 16x64 output strip (4 N-tiles),
// so each A fragment (1x b64 load per lane per K-step) feeds 4 WMMAs.
//
// Fragment layouts per CDNA5 ISA 7.12.2:
//   A (16x4 f32, 2 VGPR): lanes 0-15 hold M=lane, K={k,k+1};
//                         lanes 16-31 hold M=lane-16, K={k+2,k+3}
//   B (4x16 f32, 2 VGPR): lanes 0-15 hold N=lane, K={k,k+1};
//                         lanes 16-31 hold N=lane-16, K={k+2,k+3}
//   C/D (16x16 f32, 8 VGPR): vgpr i -> (M=i, N=lane) lanes 0-15,
//                            (M=8+i, N=lane-16) lanes 16-31
// ---------------------------------------------------------------------------
__global__ void GCN_wmma_gemm_kernel(const float* __restrict__ A,   // [M, 256]
                                     const float* __restrict__ W,   // [256, 256]
                                     const float* __restrict__ bias,// [256]
                                     float* __restrict__ out,       // [M, 256]
                                     int mTiles) {
  const int lane       = threadIdx.x & 31;
  const int waveInBlk  = threadIdx.x >> 5;
  const int globalWave = blockIdx.x * (blockDim.x >> 5) + waveInBlk;
  const int numWaves   = mTiles * (OUT_DIM / 64);   // 4 N-groups of 64 cols
  if (globalWave >= numWaves) return;   // uniform per wave: EXEC stays all-1s

  const int mTile  = globalWave >> 2;   // / 4
  const int nGroup = globalWave & 3;    // 64-column group

  const int half = lane >> 4;           // 0: lanes 0-15, 1: lanes 16-31
  const int l15  = lane & 15;

  // A row this lane reads; 8B-aligned (row base is 1KB aligned, half*2*4=8B).
  const float* aRow = A + (long long)(mTile * 16 + l15) * IN_DIM + half * 2;
  // B base: W[(k + half*2 + {0,1}) * 256 + nGroup*64 + j*16 + l15]
  const float* bBase = W + (long long)(half * 2) * OUT_DIM + nGroup * 64 + l15;

  v8f c0 = {}, c1 = {}, c2 = {}, c3 = {};

  #pragma unroll 4
  for (int k = 0; k < IN_DIM; k += 4) {
    const v2f a = *(const v2f*)(aRow + k);          // global_load_b64

    const float* bk = bBase + (long long)k * OUT_DIM;
    v2f b0, b1, b2, b3;
    b0.x = bk[0];        b0.y = bk[OUT_DIM];
    b1.x = bk[16];       b1.y = bk[OUT_DIM + 16];
    b2.x = bk[32];       b2.y = bk[OUT_DIM + 32];
    b3.x = bk[48];       b3.y = bk[OUT_DIM + 48];

    // 8 args: (neg_a, A, neg_b, B, c_mod, C, reuse_a, reuse_b)
    c0 = __builtin_amdgcn_wmma_f32_16x16x4_f32(false, a, false, b0, (short)0, c0, false, false);
    c1 = __builtin_amdgcn_wmma_f32_16x16x4_f32(false, a, false, b1, (short)0, c1, false, false);
    c2 = __builtin_amdgcn_wmma_f32_16x16x4_f32(false, a, false, b2, (short)0, c2, false, false);
    c3 = __builtin_amdgcn_wmma_f32_16x16x4_f32(false, a, false, b3, (short)0, c3, false, false);
  }

  // Epilogue: bias add + store. vgpr i of each accumulator maps to row
  // (mTile*16 + half*8 + i), column (nGroup*64 + j*16 + l15).
  float* outRow = out + (long long)(mTile * 16 + half * 8) * OUT_DIM
                      + nGroup * 64 + l15;
  const float bv0 = bias[nGroup * 64 + l15];
  const float bv1 = bias[nGroup * 64 + 16 + l15];
  const float bv2 = bias[nGroup * 64 + 32 + l15];
  const float bv3 = bias[nGroup * 64 + 48 + l15];

  #pragma unroll
  for (int i = 0; i < 8; ++i) {
    float* o = outRow + (long long)i * OUT_DIM;
    o[0]  = c0[i] + bv0;
    o[16] = c1[i] + bv1;
    o[32] = c2[i] + bv2;
    o[48] = c3[i] + bv3;
  }
}

// ---------------------------------------------------------------------------
extern "C" void kernel_launch(void* const* d_in, const int* in_sizes, int n_in,
                              void* d_out, int out_size, void* d_ws, size_t ws_size,
                              hipStream_t stream) {
  const int*   row  = (const int*)d_in[0];
  const int*   col  = (const int*)d_in[1];
  const float* vals = (const float*)d_in[2];
  const float* x    = (const float*)d_in[3];
  const float* W    = (const float*)d_in[4];
  const float* b    = (const float*)d_in[5];
  float*       out  = (float*)d_out;

  const int E = in_sizes[0];
  const int N = in_sizes[3] / IN_DIM;

  float* agg = (float*)d_ws;                       // [N, IN_DIM] scratch
  const long long agg4 = ((long long)N * IN_DIM) / 4;

  // Phase 0: zero agg (b128 stores, grid-stride)
  GCN_zero_kernel<<<2048, 256, 0, stream>>>((float4*)agg, agg4);

  // Phase 1: scatter-add SpMM (one wave32 per edge, 8 edges per block)
  {
    const int wavesPerBlock = 8;                   // 256 threads
    const int blocks = (E + wavesPerBlock - 1) / wavesPerBlock;
    GCN_spmm_scatter_kernel<<<blocks, wavesPerBlock * 32, 0, stream>>>(
        row, col, vals, x, agg, E);
  }

  // Phase 2: WMMA GEMM + bias (one wave per 16x64 output strip)
  {
    const int mTiles        = (N + 15) / 16;       // 6250 for N=100000 (exact)
    const int numWaves      = mTiles * (OUT_DIM / 64);
    const int wavesPerBlock = 8;                   // 256 threads = 8 waves
    const int blocks        = (numWaves + wavesPerBlock - 1) / wavesPerBlock;
    GCN_wmma_gemm_kernel<<<blocks, wavesPerBlock * 32, 0, stream>>>(
        agg, W, b, out, mTiles);
  }
}